// GCNEncoder_51376398795255
// MI455X (gfx1250) — compile-verified
//
#include <hip/hip_runtime.h>

typedef float v2f __attribute__((ext_vector_type(2)));
typedef float v8f __attribute__((ext_vector_type(8)));

#define D 128

// ---- degree / normalization ------------------------------------------------
__global__ __launch_bounds__(256) void k_init_deg(float* deg, int N) {
  int i = blockIdx.x * 256 + threadIdx.x;
  if (i < N) deg[i] = 1.0f;              // self-loop contributes 1
}

__global__ __launch_bounds__(256) void k_edge_deg(const int* __restrict__ dst,
                                                  float* deg, int E) {
  int e = blockIdx.x * 256 + threadIdx.x;
  if (e < E) atomicAdd(&deg[dst[e]], 1.0f);
}

__global__ __launch_bounds__(256) void k_rsqrt(float* deg, int N) {
  int i = blockIdx.x * 256 + threadIdx.x;
  if (i < N) deg[i] = rsqrtf(deg[i]);    // deg >= 1 always (self-loop)
}

// ---- XW = X[N,128] @ W[128,128] via V_WMMA_F32_16X16X4_F32 -----------------
// One wave32 computes TWO 16x16 output tiles (rows row0..row0+15 and
// row0+16..row0+31, same 16-col slice) sharing every B (W) load between two
// WMMAs. 8 waves/block -> 32 rows x 128 cols per block.
// A (16x4, MxK): lane m=lane&15 holds row m; lanes 0-15 K={0,1}, lanes 16-31 K={2,3}.
// B (4x16, KxN): mirrored striping across lanes.
// C/D: VGPR r -> M=r (lanes 0-15) / M=r+8 (lanes 16-31), N=lane&15.
__global__ __launch_bounds__(256) void k_gemm(const float* __restrict__ X,
                                              const float* __restrict__ W,
                                              float* __restrict__ XW, int N) {
  const int row0 = blockIdx.x * 32;
  const int wave = threadIdx.x >> 5;
  const int lane = threadIdx.x & 31;
  const int col0 = wave * 16;
  const int m    = lane & 15;
  const int kh   = (lane >> 4) << 1;     // 0 or 2

  int r0 = row0 + m;       if (r0 >= N) r0 = N - 1;   // clamp loads
  int r1 = row0 + 16 + m;  if (r1 >= N) r1 = N - 1;
  const float* x0 = X + (size_t)r0 * D;
  const float* x1 = X + (size_t)r1 * D;

  v8f acc0 = {}, acc1 = {};
  for (int k = 0; k < D; k += 4) {
    v2f b, a0, a1;
    b.x  = W[(size_t)(k + kh + 0) * D + col0 + m];
    b.y  = W[(size_t)(k + kh + 1) * D + col0 + m];
    a0.x = x0[k + kh + 0];
    a0.y = x0[k + kh + 1];
    a1.x = x1[k + kh + 0];
    a1.y = x1[k + kh + 1];
    acc0 = __builtin_amdgcn_wmma_f32_16x16x4_f32(
        false, a0, false, b, (short)0, acc0, false, false);
    acc1 = __builtin_amdgcn_wmma_f32_16x16x4_f32(
        false, a1, false, b, (short)0, acc1, false, false);
  }

  const int mo  = (lane >> 4) << 3;      // 0 or 8
  float* o0 = XW + (size_t)(row0 + mo) * D + col0 + m;
  float* o1 = o0 + (size_t)16 * D;

  if (row0 + 32 <= N) {                  // fast path: full block, no guards
#pragma unroll
    for (int r = 0; r < 8; ++r) {
      o0[(size_t)r * D] = acc0[r];
      o1[(size_t)r * D] = acc1[r];
    }
  } else {                               // tail block: per-row guard
#pragma unroll
    for (int r = 0; r < 8; ++r) {
      int rr0 = row0 + mo + r;
      int rr1 = rr0 + 16;
      if (rr0 < N) o0[(size_t)r * D] = acc0[r];
      if (rr1 < N) o1[(size_t)r * D] = acc1[r];
    }
  }
}

// ---- agg[i] = dinv[i]^2 * xw[i]  (self-loop term; also initializes agg) ----
__global__ __launch_bounds__(256) void k_self_init(const float* __restrict__ xw,
                                                   const float* __restrict__ dinv,
                                                   float* __restrict__ agg, int N) {
  int t = blockIdx.x * 256 + threadIdx.x;     // N*32 threads, float4 each
  int node = t >> 5;
  if (node >= N) return;
  int lane = t & 31;
  float s = dinv[node]; s = s * s;
  float4 v = ((const float4*)(xw + (size_t)node * D))[lane];
  float4 o; o.x = v.x * s; o.y = v.y * s; o.z = v.z * s; o.w = v.w * s;
  ((float4*)(agg + (size_t)node * D))[lane] = o;
}

// ---- edge scatter: agg[dst] += dinv[src]*dinv[dst] * xw[src] ---------------
// Grid-stride, one wave32 per edge per iteration; each lane moves a float4
// (4 atomic f32 adds). The next iteration's source row is prefetched
// (global_prefetch_b8) to hide the random-gather latency behind the atomics.
__global__ __launch_bounds__(256) void k_edge_scatter(const int* __restrict__ src,
                                                      const int* __restrict__ dst,
                                                      const float* __restrict__ dinv,
                                                      const float* __restrict__ xw,
                                                      float* __restrict__ agg, int E) {
  const int lane   = threadIdx.x & 31;
  const int waveId = blockIdx.x * 8 + (threadIdx.x >> 5);
  const int stride = gridDim.x * 8;

  for (int e = waveId; e < E; e += stride) {
    int e2 = e + stride;                       // prefetch next edge's row
    if (e2 < E) {
      int s2 = src[e2];
      __builtin_prefetch(xw + (size_t)s2 * D + lane * 4, 0, 3);
    }
    int s = src[e], d = dst[e];
    float norm = dinv[s] * dinv[d];
    float4 v = ((const float4*)(xw + (size_t)s * D))[lane];
    float* ar = agg + (size_t)d * D + lane * 4;
    atomicAdd(ar + 0, norm * v.x);
    atomicAdd(ar + 1, norm * v.y);
    atomicAdd(ar + 2, norm * v.z);
    atomicAdd(ar + 3, norm * v.w);
  }
}

// ---- relu(agg + b) then accumulate per-graph sums + counts -----------------
__global__ __launch_bounds__(256) void k_pool(const float* __restrict__ agg,
                                              const float* __restrict__ bias,
                                              const int* __restrict__ batch,
                                              float* __restrict__ pool,
                                              float* __restrict__ cnt, int N) {
  int t = blockIdx.x * 256 + threadIdx.x;     // N*32 threads, float4 each
  int node = t >> 5;
  if (node >= N) return;
  int lane = t & 31;
  int g = batch[node];
  float4 v = ((const float4*)(agg + (size_t)node * D))[lane];
  float4 bb = ((const float4*)bias)[lane];
  v.x = fmaxf(v.x + bb.x, 0.0f);
  v.y = fmaxf(v.y + bb.y, 0.0f);
  v.z = fmaxf(v.z + bb.z, 0.0f);
  v.w = fmaxf(v.w + bb.w, 0.0f);
  float* pr = pool + (size_t)g * D + lane * 4;
  atomicAdd(pr + 0, v.x);
  atomicAdd(pr + 1, v.y);
  atomicAdd(pr + 2, v.z);
  atomicAdd(pr + 3, v.w);
  if (lane == 0) atomicAdd(&cnt[g], 1.0f);
}

__global__ __launch_bounds__(256) void k_div(const float* __restrict__ pool,
                                             const float* __restrict__ cnt,
                                             float* __restrict__ out, int G) {
  int t = blockIdx.x * 256 + threadIdx.x;     // G*128
  if (t >= G * D) return;
  int g = t >> 7;
  out[t] = pool[t] / fmaxf(cnt[g], 1.0f);
}

// ---------------------------------------------------------------------------
extern "C" void kernel_launch(void* const* d_in, const int* in_sizes, int n_in,
                              void* d_out, int out_size, void* d_ws, size_t ws_size,
                              hipStream_t stream) {
  const float* x     = (const float*)d_in[0];   // [N,128]
  const int*   ei    = (const int*)d_in[1];     // [2,E]
  const int*   batch = (const int*)d_in[2];     // [N]
  const float* Wm    = (const float*)d_in[3];   // [128,128]
  const float* bias  = (const float*)d_in[4];   // [128]

  const int N = in_sizes[0] / D;
  const int E = in_sizes[1] / 2;
  const int G = out_size / D;
  const int* src = ei;
  const int* dst = ei + E;

  float* ws = (float*)d_ws;
  size_t nd = (size_t)N * D;
  float* xw   = ws;                              // N*128
  float* agg  = ws + nd;                         // N*128
  float* dinv = ws + 2 * nd;                     // N
  size_t dv_pad = (((size_t)N + 3) & ~(size_t)3);
  float* pool = dinv + dv_pad;                   // G*128
  float* cnt  = pool + (size_t)G * D;            // G

  // zero the pool accumulators (+counts), contiguous region
  hipMemsetAsync(pool, 0, ((size_t)G * D + G) * sizeof(float), stream);

  k_init_deg  <<<(N + 255) / 256, 256, 0, stream>>>(dinv, N);
  k_edge_deg  <<<(E + 255) / 256, 256, 0, stream>>>(dst, dinv, E);
  k_rsqrt     <<<(N + 255) / 256, 256, 0, stream>>>(dinv, N);
  k_gemm      <<<(N + 31) / 32,   256, 0, stream>>>(x, Wm, xw, N);
  k_self_init <<<(int)(((size_t)N * 32 + 255) / 256), 256, 0, stream>>>(xw, dinv, agg, N);

  int sc_blocks = (E + 7) / 8;
  if (sc_blocks > 8192) sc_blocks = 8192;        // grid-stride wave-per-edge
  k_edge_scatter<<<sc_blocks, 256, 0, stream>>>(src, dst, dinv, xw, agg, E);

  k_pool <<<(int)(((size_t)N * 32 + 255) / 256), 256, 0, stream>>>(agg, bias, batch, pool, cnt, N);
  k_div  <<<(int)(((size_t)G * D + 255) / 256), 256, 0, stream>>>(pool, cnt, (float*)d_out, G);
}